// SMPL_63350767616280
// MI455X (gfx1250) — compile-verified
//
#include <hip/hip_runtime.h>
#include <hip/hip_bf16.h>
#include <math.h>

#define NVV 100000
#define NJJ 24

typedef float v2f __attribute__((ext_vector_type(2)));
typedef float v8f __attribute__((ext_vector_type(8)));

// ---------------------------------------------------------------------------
// K0: v_shaped[v,c] = v_template[v,c] + shapedirs[v,c,:] . betas
// ---------------------------------------------------------------------------
__global__ __launch_bounds__(256) void smpl_vshaped(
    const float* __restrict__ v_template,
    const float* __restrict__ shapedirs,
    const float* __restrict__ betas,
    float* __restrict__ vsh)
{
    int i = blockIdx.x * 256 + threadIdx.x;          // over NV*3
    if (i >= NVV * 3) return;
    const float* row = shapedirs + (size_t)i * 10;
    float acc = v_template[i];
#pragma unroll
    for (int b = 0; b < 10; ++b) acc = fmaf(row[b], betas[b], acc);
    vsh[i] = acc;
}

// ---------------------------------------------------------------------------
// K1: Jloc[j,:] = sum_v J_regressor[j,v] * v_shaped[v,:]
// one block per joint; deterministic LDS tree reduction
// J_regressor is a 9.6MB single-use stream -> non-temporal loads
// ---------------------------------------------------------------------------
__global__ __launch_bounds__(256) void smpl_jloc(
    const float* __restrict__ Jreg,
    const float* __restrict__ vsh,
    const float* __restrict__ v_template,
    const float* __restrict__ shapedirs,
    const float* __restrict__ betas,
    int use_vsh,
    float* __restrict__ ws_jloc)
{
    __shared__ float s0[256], s1[256], s2[256];
    __shared__ float bsh[10];
    const int tid = threadIdx.x;
    const int j = blockIdx.x;
    if (tid < 10) bsh[tid] = betas[tid];
    __syncthreads();

    float a0 = 0.f, a1 = 0.f, a2 = 0.f;
    const float* jrow = Jreg + (size_t)j * NVV;
    for (int v = tid; v < NVV; v += 256) {
        float w = __builtin_nontemporal_load(&jrow[v]);
        float x, y, z;
        if (use_vsh) {
            x = vsh[v * 3 + 0]; y = vsh[v * 3 + 1]; z = vsh[v * 3 + 2];
        } else {
            const float* sd = shapedirs + (size_t)v * 30;
            x = v_template[v * 3 + 0];
            y = v_template[v * 3 + 1];
            z = v_template[v * 3 + 2];
#pragma unroll
            for (int b = 0; b < 10; ++b) {
                x = fmaf(sd[b],      bsh[b], x);
                y = fmaf(sd[10 + b], bsh[b], y);
                z = fmaf(sd[20 + b], bsh[b], z);
            }
        }
        a0 = fmaf(w, x, a0); a1 = fmaf(w, y, a1); a2 = fmaf(w, z, a2);
    }
    s0[tid] = a0; s1[tid] = a1; s2[tid] = a2;
    __syncthreads();
    for (int off = 128; off > 0; off >>= 1) {
        if (tid < off) {
            s0[tid] += s0[tid + off];
            s1[tid] += s1[tid + off];
            s2[tid] += s2[tid + off];
        }
        __syncthreads();
    }
    if (tid == 0) {
        ws_jloc[j * 3 + 0] = s0[0];
        ws_jloc[j * 3 + 1] = s1[0];
        ws_jloc[j * 3 + 2] = s2[0];
    }
}

// ---------------------------------------------------------------------------
// K2: Rodrigues, lrotmin, kinematic chain (by tree level), Grel -> GrelT[24][16]
// single wave
// ---------------------------------------------------------------------------
__global__ __launch_bounds__(32) void smpl_chain(
    const float* __restrict__ pose,
    const float* __restrict__ ws_jloc,
    float* __restrict__ ws_lrot,
    float* __restrict__ ws_grelT)
{
    __shared__ float R[24][9];
    __shared__ float G[24][12];   // rows 0..2 of 4x4, row-major 3x4 (bottom row implied)
    const int t = threadIdx.x;

    if (t < 24) {
        float rx = pose[t * 3 + 0], ry = pose[t * 3 + 1], rz = pose[t * 3 + 2];
        float th = sqrtf(rx * rx + ry * ry + rz * rz) + 1e-8f;
        float inv = 1.0f / th;
        float x = rx * inv, y = ry * inv, z = rz * inv;
        float c = cosf(th), s = sinf(th), ic = 1.0f - c;
        R[t][0] = c + ic * x * x;     R[t][1] = ic * x * y - s * z; R[t][2] = ic * x * z + s * y;
        R[t][3] = ic * x * y + s * z; R[t][4] = c + ic * y * y;     R[t][5] = ic * y * z - s * x;
        R[t][6] = ic * x * z - s * y; R[t][7] = ic * y * z + s * x; R[t][8] = c + ic * z * z;
    }
    __syncthreads();

    for (int p = t; p < 207; p += 32) {
        int j = 1 + p / 9, e = p % 9;
        float id = (e == 0 || e == 4 || e == 8) ? 1.0f : 0.0f;
        ws_lrot[p] = R[j][e] - id;
    }

    const int parent[24] = {0,0,0,0,1,2,3,4,5,6,7,8,9,9,9,12,13,14,16,17,18,19,20,21};
    const int level[24]  = {0,1,1,1,2,2,2,3,3,3,4,4,4,4,4,5,5,5,6,6,7,7,8,8};

    for (int lvl = 0; lvl < 9; ++lvl) {
        if (t < 24 && level[t] == lvl) {
            int p = parent[t];
            float tx = ws_jloc[t * 3 + 0], ty = ws_jloc[t * 3 + 1], tz = ws_jloc[t * 3 + 2];
            if (lvl == 0) {
#pragma unroll
                for (int r = 0; r < 3; ++r) {
                    G[t][r * 4 + 0] = R[t][r * 3 + 0];
                    G[t][r * 4 + 1] = R[t][r * 3 + 1];
                    G[t][r * 4 + 2] = R[t][r * 3 + 2];
                }
                G[t][3] = tx; G[t][7] = ty; G[t][11] = tz;
            } else {
                tx -= ws_jloc[p * 3 + 0]; ty -= ws_jloc[p * 3 + 1]; tz -= ws_jloc[p * 3 + 2];
#pragma unroll
                for (int r = 0; r < 3; ++r) {
                    float g0 = G[p][r * 4 + 0], g1 = G[p][r * 4 + 1];
                    float g2 = G[p][r * 4 + 2], g3 = G[p][r * 4 + 3];
                    G[t][r * 4 + 0] = g0 * R[t][0] + g1 * R[t][3] + g2 * R[t][6];
                    G[t][r * 4 + 1] = g0 * R[t][1] + g1 * R[t][4] + g2 * R[t][7];
                    G[t][r * 4 + 2] = g0 * R[t][2] + g1 * R[t][5] + g2 * R[t][8];
                    G[t][r * 4 + 3] = g0 * tx + g1 * ty + g2 * tz + g3;
                }
            }
        }
        __syncthreads();
    }

    if (t < 24) {
        float jx = ws_jloc[t * 3 + 0], jy = ws_jloc[t * 3 + 1], jz = ws_jloc[t * 3 + 2];
#pragma unroll
        for (int r = 0; r < 3; ++r) {
            float gj = G[t][r * 4 + 0] * jx + G[t][r * 4 + 1] * jy + G[t][r * 4 + 2] * jz;
            ws_grelT[t * 16 + r * 4 + 0] = G[t][r * 4 + 0];
            ws_grelT[t * 16 + r * 4 + 1] = G[t][r * 4 + 1];
            ws_grelT[t * 16 + r * 4 + 2] = G[t][r * 4 + 2];
            ws_grelT[t * 16 + r * 4 + 3] = G[t][r * 4 + 3] - gj;
        }
        ws_grelT[t * 16 + 12] = 0.f; ws_grelT[t * 16 + 13] = 0.f;
        ws_grelT[t * 16 + 14] = 0.f; ws_grelT[t * 16 + 15] = 0.f;
    }
}

// ---------------------------------------------------------------------------
// K3: main per-vertex kernel. 256 threads = 8 waves; 1 vertex per lane.
//  - stream shapedirs/posedirs rows (lrotmin broadcast from LDS)
//    posedirs (248MB > 192MB L2, single-use) -> non-temporal loads
//  - skinning T = W(16x24) x GrelT(24x16) via V_WMMA_F32_16X16X4_F32
// ---------------------------------------------------------------------------
__global__ __launch_bounds__(256) void smpl_main(
    const float* __restrict__ v_template,
    const float* __restrict__ shapedirs,
    const float* __restrict__ posedirs,
    const float* __restrict__ skinw,
    const float* __restrict__ betas,
    const float* __restrict__ lrot_g,
    const float* __restrict__ grelT_g,
    float* __restrict__ out)
{
    __shared__ float lrot[207];
    __shared__ float betas_s[10];
    __shared__ float grelT[24 * 16];
    __shared__ float wlds[8][32 * 25];   // per-wave skin weights, padded stride 25
    __shared__ float tlds[8][32 * 17];   // per-wave T rows, padded stride 17

    const int tid = threadIdx.x;
    for (int i = tid; i < 207; i += 256) lrot[i] = lrot_g[i];
    for (int i = tid; i < 384; i += 256) grelT[i] = grelT_g[i];
    if (tid < 10) betas_s[tid] = betas[tid];
    __syncthreads();

    const int wave = tid >> 5;
    const int lane = tid & 31;
    const int v = blockIdx.x * 256 + tid;
    const int vc = (v < NVV) ? v : (NVV - 1);   // clamp: keep EXEC all-ones for WMMA

    // v_shaped
    float vs0, vs1, vs2;
    {
        const float* sd = shapedirs + (size_t)vc * 30;
        vs0 = v_template[vc * 3 + 0];
        vs1 = v_template[vc * 3 + 1];
        vs2 = v_template[vc * 3 + 2];
#pragma unroll
        for (int b = 0; b < 10; ++b) {
            float bb = betas_s[b];
            vs0 = fmaf(sd[b],      bb, vs0);
            vs1 = fmaf(sd[10 + b], bb, vs1);
            vs2 = fmaf(sd[20 + b], bb, vs2);
        }
    }

    // v_posed: 3 dots of length 207 against lrotmin (dominant 248MB stream, NT)
    float p0 = 0.f, p1 = 0.f, p2 = 0.f;
    {
        const float* row = posedirs + (size_t)vc * 621;
#pragma unroll 9
        for (int k = 0; k < 207; ++k) {
            float l = lrot[k];
            p0 = fmaf(__builtin_nontemporal_load(&row[k]),       l, p0);
            p1 = fmaf(__builtin_nontemporal_load(&row[207 + k]), l, p1);
            p2 = fmaf(__builtin_nontemporal_load(&row[414 + k]), l, p2);
        }
    }
    const float vx = vs0 + p0, vy = vs1 + p1, vz = vs2 + p2;

    // stage this lane's 24 skin weights into LDS (wave-private slice)
    {
        const float* w = skinw + (size_t)vc * 24;
        float* wl = &wlds[wave][lane * 25];
#pragma unroll
        for (int j = 0; j < 24; ++j) wl[j] = __builtin_nontemporal_load(&w[j]);
    }
    __syncthreads();

    // WMMA skinning: D(16x16) = A(16x24 weights) x B(24x16 GrelT), K chunked by 4
    const int half = lane >> 4;     // 0: K rows {0,1}; 1: K rows {2,3} of each chunk
    const int l16  = lane & 15;
    v8f acc0 = {}; v8f acc1 = {};
#pragma unroll
    for (int kc = 0; kc < 6; ++kc) {
        const int k0 = kc * 4 + half * 2;
        v2f b;
        b.x = grelT[(k0 + 0) * 16 + l16];
        b.y = grelT[(k0 + 1) * 16 + l16];
        v2f a0, a1;
        a0.x = wlds[wave][l16 * 25 + k0 + 0];
        a0.y = wlds[wave][l16 * 25 + k0 + 1];
        a1.x = wlds[wave][(16 + l16) * 25 + k0 + 0];
        a1.y = wlds[wave][(16 + l16) * 25 + k0 + 1];
        acc0 = __builtin_amdgcn_wmma_f32_16x16x4_f32(false, a0, false, b, (short)0, acc0, false, false);
        acc1 = __builtin_amdgcn_wmma_f32_16x16x4_f32(false, a1, false, b, (short)0, acc1, false, false);
    }

    // scatter D through LDS so each lane recovers its own vertex's T row
    {
        float* tl = &tlds[wave][0];
#pragma unroll
        for (int r = 0; r < 8; ++r) {
            const int m0 = r + half * 8;
            tl[m0 * 17 + l16]        = acc0[r];   // tile 0: vertices 0..15
            tl[(16 + m0) * 17 + l16] = acc1[r];   // tile 1: vertices 16..31
        }
    }
    __syncthreads();

    float T[12];
#pragma unroll
    for (int n = 0; n < 12; ++n) T[n] = tlds[wave][lane * 17 + n];

    const float ox = T[0] * vx + T[1]  * vy + T[2]  * vz + T[3];
    const float oy = T[4] * vx + T[5]  * vy + T[6]  * vz + T[7];
    const float oz = T[8] * vx + T[9]  * vy + T[10] * vz + T[11];

    if (v < NVV) {
        __builtin_nontemporal_store(ox, &out[v * 3 + 0]);
        __builtin_nontemporal_store(oy, &out[v * 3 + 1]);
        __builtin_nontemporal_store(oz, &out[v * 3 + 2]);
    }
}

// ---------------------------------------------------------------------------
extern "C" void kernel_launch(void* const* d_in, const int* in_sizes, int n_in,
                              void* d_out, int out_size, void* d_ws, size_t ws_size,
                              hipStream_t stream) {
    const float* betas      = (const float*)d_in[0];
    const float* pose       = (const float*)d_in[1];
    const float* v_template = (const float*)d_in[2];
    const float* Jreg       = (const float*)d_in[3];
    const float* skinw      = (const float*)d_in[4];
    const float* shapedirs  = (const float*)d_in[5];
    const float* posedirs   = (const float*)d_in[6];
    float* out = (float*)d_out;
    float* ws  = (float*)d_ws;

    float* ws_jloc  = ws;          // 72 floats
    float* ws_lrot  = ws + 72;     // 207 floats
    float* ws_grelT = ws + 288;    // 384 floats
    float* ws_vsh   = ws + 1024;   // NV*3 floats (fast path only)

    const int use_vsh = (ws_size >= (size_t)(1024 + NVV * 3) * sizeof(float)) ? 1 : 0;

    if (use_vsh) {
        smpl_vshaped<<<(NVV * 3 + 255) / 256, 256, 0, stream>>>(
            v_template, shapedirs, betas, ws_vsh);
    }
    smpl_jloc<<<24, 256, 0, stream>>>(
        Jreg, ws_vsh, v_template, shapedirs, betas, use_vsh, ws_jloc);
    smpl_chain<<<1, 32, 0, stream>>>(pose, ws_jloc, ws_lrot, ws_grelT);
    smpl_main<<<(NVV + 255) / 256, 256, 0, stream>>>(
        v_template, shapedirs, posedirs, skinw, betas, ws_lrot, ws_grelT, out);
}